// ToZ_17566416240900
// MI455X (gfx1250) — compile-verified
//
#include <hip/hip_runtime.h>
#include <stdint.h>

// ToZ forward for x:(1,3,64,64) f32.
//   out[0, :]   = x                           (12288 floats)
//   out[1+i, j] = (i == j) ? EPS : 0          (12288 x 12288 diagonal block)
//
// Pure store-bandwidth problem (~604 MB written once, 0 FLOPs; ~26 us at
// 23.3 TB/s). The zero-fill is streamed with CDNA5 async LDS->global stores:
// a 512 B zero tile lives in LDS, and GLOBAL_STORE_ASYNC_FROM_LDS_B128
// (ASYNCcnt-tracked) broadcasts it to memory with th:TH_STORE_NT so the
// single-use output does not thrash the 192 MB L2. Each thread waits on its
// own ASYNCcnt before the single EPS diagonal overwrite (async vs. normal
// stores are unordered across types).

typedef float v4f __attribute__((ext_vector_type(4)));

#define TOZ_EPS 0.01f

__global__ __launch_bounds__(256) void toz_kernel(const float* __restrict__ x,
                                                  float* __restrict__ out,
                                                  int row_len) {
  __shared__ __align__(16) float lds_zero[128];  // 512 B zero tile (16 B/lane)

  const int tid  = threadIdx.x;
  const int row  = blockIdx.x;                   // 0 .. pad  (12289 rows)
  const int nvec = row_len >> 2;                 // float4 vectors per row
  float* __restrict__ rowp = out + (size_t)row * (size_t)row_len;

  if (row == 0) {
    // Copy x into the first row (48 KB total, negligible traffic).
    const v4f* __restrict__ src = (const v4f*)x;
    v4f* __restrict__ dst = (v4f*)rowp;
    for (int v = tid; v < nvec; v += blockDim.x) {
      v4f val = src[v];
      __builtin_nontemporal_store(val, dst + v);
    }
    return;
  }

  // Stage 512 B of zeros in LDS (one float4 per lane of wave 0).
  if (tid < 32) {
    ((v4f*)lds_zero)[tid] = (v4f){0.0f, 0.0f, 0.0f, 0.0f};
  }
  __syncthreads();

  // LDS byte offset for this lane (low 32 bits of the generic address are the
  // LDS aperture offset per the CDNA5 flat-address mapping).
  const unsigned lds_off =
      (unsigned)(uintptr_t)(const void*)&lds_zero[(tid & 31) * 4];

  const uint64_t row_base = (uint64_t)(uintptr_t)rowp;

  // Bulk zero-fill: fire-and-forget async streaming stores from LDS.
  for (int v = tid; v < nvec; v += blockDim.x) {
    uint64_t gaddr = row_base + (uint64_t)v * 16u;
    asm volatile("global_store_async_from_lds_b128 %0, %1, off th:TH_STORE_NT"
                 :
                 : "v"(gaddr), "v"(lds_off)
                 : "memory");
  }

  // Drain this wave's async stores before touching the diagonal element.
  asm volatile("s_wait_asynccnt 0x0" ::: "memory");

  // Diagonal entry: out[row, row-1] = EPS, written by the thread whose async
  // store covered that float4 (same wave -> ordering via ASYNCcnt wait).
  const int d  = row - 1;
  const int dv = d >> 2;
  if (d < row_len && tid == (dv % (int)blockDim.x)) {
    rowp[d] = TOZ_EPS;
  }
}

extern "C" void kernel_launch(void* const* d_in, const int* in_sizes, int n_in,
                              void* d_out, int out_size, void* d_ws, size_t ws_size,
                              hipStream_t stream) {
  const float* x = (const float*)d_in[0];
  float* out     = (float*)d_out;

  // x is (K=1, 3, 64, 64) flattened: feature length == in_sizes[0] == 12288.
  const int row_len = in_sizes[0];          // 12288
  const int rows    = out_size / row_len;   // 12289 = K + pad

  toz_kernel<<<dim3(rows), dim3(256), 0, stream>>>(x, out, row_len);
}